// GATLayer_69844758167586
// MI455X (gfx1250) — compile-verified
//
#include <hip/hip_runtime.h>
#include <math.h>

#define HEADS     4
#define HEAD_DIM  32
#define IN_CH     128
#define OUT_CH    128   // HEADS * HEAD_DIM
#define NEG_SLOPE 0.2f

typedef float v2f __attribute__((ext_vector_type(2)));
typedef float v8f __attribute__((ext_vector_type(8)));

// ---------------------------------------------------------------------------
// Float atomic-max via integer atomics (native on gfx1250; handles mixed sign
// correctly given -inf initialization).
// ---------------------------------------------------------------------------
__device__ __forceinline__ void atomic_max_float(float* addr, float val) {
    if (val >= 0.0f) {
        atomicMax((int*)addr, __float_as_int(val));
    } else {
        atomicMin((unsigned int*)addr, __float_as_uint(val));
    }
}

// ---------------------------------------------------------------------------
// 0) Init: zero the output accumulator, seg_sum; seg_max = -inf.
// ---------------------------------------------------------------------------
__global__ __launch_bounds__(256) void gat_init(float* __restrict__ out,
                                                float* __restrict__ seg_max,
                                                float* __restrict__ seg_sum,
                                                int N) {
    int i = blockIdx.x * 256 + threadIdx.x;
    if (i < N * OUT_CH) out[i] = 0.0f;
    if (i < N * HEADS) {
        seg_max[i] = -INFINITY;
        seg_sum[i] = 0.0f;
    }
}

// ---------------------------------------------------------------------------
// 1) GEMM: xt[N,128] = x[N,128] @ W[128,128] via V_WMMA_F32_16X16X4_F32.
//    Block = 256 threads = 8 waves; block b owns 16 rows, wave w owns cols
//    [16w, 16w+16). K loop = 32 WMMA steps of K=4.
//    A frag layout: lane L (m = L&15, khalf = L>>4): a = {K=k+2*khalf, K=k+2*khalf+1}
//    B frag layout mirrors A's K distribution; n = L&15.
//    D layout: col n = L&15, row m = v + 8*(L>>4) for VGPR v of 8.
//    Store epilogue: unguarded fast path when the full 16-row tile is in
//    range (always, for N % 16 == 0) -> no exec-mask branching per row.
// ---------------------------------------------------------------------------
__global__ __launch_bounds__(256) void gat_gemm(const float* __restrict__ x,
                                                const float* __restrict__ W,
                                                float* __restrict__ xt,
                                                int N) {
    const int lane  = threadIdx.x & 31;
    const int wave  = threadIdx.x >> 5;       // 0..7 -> column tile
    const int row0  = blockIdx.x * 16;
    const int col0  = wave * 16;
    const int mrow  = lane & 15;              // A row / B,D column index
    const int khalf = lane >> 4;              // 0 or 1

    int arow = row0 + mrow;
    if (arow >= N) arow = N - 1;              // clamp; stores are bounds-checked
    const float* aptr = x + (size_t)arow * IN_CH + khalf * 2;
    const float* bptr = W + (size_t)(khalf * 2) * OUT_CH + col0 + mrow;

    v8f c = {};
#pragma unroll 4
    for (int k = 0; k < IN_CH; k += 4) {
        v2f a, b;
        a.x = aptr[k + 0];                      // K = k + 2*khalf
        a.y = aptr[k + 1];                      // K = k + 2*khalf + 1
        b.x = bptr[(size_t)k * OUT_CH];         // W[k + 2*khalf      ][col]
        b.y = bptr[(size_t)(k + 1) * OUT_CH];   // W[k + 2*khalf + 1  ][col]
        // (neg_a, A, neg_b, B, c_mod, C, reuse_a, reuse_b)
        c = __builtin_amdgcn_wmma_f32_16x16x4_f32(false, a, false, b,
                                                  (short)0, c, false, false);
    }

    const int ncol  = lane & 15;
    const int mbase = (lane >> 4) * 8;
    float* dptr = xt + (size_t)(row0 + mbase) * OUT_CH + col0 + ncol;
    if (row0 + 16 <= N) {
        // Fast path: full tile in range; 8 back-to-back stores (clauseable).
#pragma unroll
        for (int v = 0; v < 8; ++v) {
            dptr[(size_t)v * OUT_CH] = c[v];
        }
    } else {
        // Ragged final tile (not hit when N % 16 == 0).
#pragma unroll
        for (int v = 0; v < 8; ++v) {
            if (row0 + mbase + v < N) dptr[(size_t)v * OUT_CH] = c[v];
        }
    }
}

// ---------------------------------------------------------------------------
// 2) Per-node attention dots: one wave per node, shfl_xor tree reduction.
//    a_src[n,h] = sum_c xt[n,h,c]*att_src[h,c]  (same for dst).
// ---------------------------------------------------------------------------
__global__ __launch_bounds__(256) void gat_att_dots(const float* __restrict__ xt,
                                                    const float* __restrict__ att_src,
                                                    const float* __restrict__ att_dst,
                                                    float* __restrict__ a_src,
                                                    float* __restrict__ a_dst,
                                                    int N) {
    int gw   = (blockIdx.x * 256 + threadIdx.x) >> 5;  // node id
    int lane = threadIdx.x & 31;
    if (gw >= N) return;
    const float* row = xt + (size_t)gw * OUT_CH;
#pragma unroll
    for (int h = 0; h < HEADS; ++h) {
        float xv = row[h * HEAD_DIM + lane];
        float s  = xv * att_src[h * HEAD_DIM + lane];
        float d  = xv * att_dst[h * HEAD_DIM + lane];
#pragma unroll
        for (int off = 16; off > 0; off >>= 1) {
            s += __shfl_xor(s, off, 32);
            d += __shfl_xor(d, off, 32);
        }
        if (lane == 0) {
            a_src[(size_t)gw * HEADS + h] = s;
            a_dst[(size_t)gw * HEADS + h] = d;
        }
    }
}

// ---------------------------------------------------------------------------
// 3) Edge logits: alpha = LeakyReLU(a_src[s]+a_dst[d]); atomic seg-max on dst.
//    Edges e < E come from edge_index; e >= E are the self-loops.
// ---------------------------------------------------------------------------
__global__ __launch_bounds__(256) void gat_edge_logits(const int* __restrict__ ei,
                                                       int E, int N,
                                                       const float* __restrict__ a_src,
                                                       const float* __restrict__ a_dst,
                                                       float* __restrict__ alpha,
                                                       float* __restrict__ seg_max) {
    int e = blockIdx.x * 256 + threadIdx.x;
    int T = E + N;
    if (e >= T) return;
    int s, d;
    if (e < E) { s = ei[e]; d = ei[E + e]; } else { s = d = e - E; }

    float4 as = *(const float4*)(a_src + (size_t)s * HEADS);
    float4 ad = *(const float4*)(a_dst + (size_t)d * HEADS);
    float4 al;
    al.x = as.x + ad.x; al.x = (al.x >= 0.0f) ? al.x : NEG_SLOPE * al.x;
    al.y = as.y + ad.y; al.y = (al.y >= 0.0f) ? al.y : NEG_SLOPE * al.y;
    al.z = as.z + ad.z; al.z = (al.z >= 0.0f) ? al.z : NEG_SLOPE * al.z;
    al.w = as.w + ad.w; al.w = (al.w >= 0.0f) ? al.w : NEG_SLOPE * al.w;
    *(float4*)(alpha + (size_t)e * HEADS) = al;

    float* mp = seg_max + (size_t)d * HEADS;
    atomic_max_float(mp + 0, al.x);
    atomic_max_float(mp + 1, al.y);
    atomic_max_float(mp + 2, al.z);
    atomic_max_float(mp + 3, al.w);
}

// ---------------------------------------------------------------------------
// 4) Edge exp + segment-sum: alpha = exp(alpha - seg_max[d]); atomic-add sums.
// ---------------------------------------------------------------------------
__global__ __launch_bounds__(256) void gat_edge_exp(const int* __restrict__ ei,
                                                    int E, int N,
                                                    const float* __restrict__ seg_max,
                                                    float* __restrict__ alpha,
                                                    float* __restrict__ seg_sum) {
    int e = blockIdx.x * 256 + threadIdx.x;
    int T = E + N;
    if (e >= T) return;
    int d = (e < E) ? ei[E + e] : (e - E);

    float4 al = *(float4*)(alpha + (size_t)e * HEADS);
    float4 mx = *(const float4*)(seg_max + (size_t)d * HEADS);
    al.x = expf(al.x - mx.x);
    al.y = expf(al.y - mx.y);
    al.z = expf(al.z - mx.z);
    al.w = expf(al.w - mx.w);
    *(float4*)(alpha + (size_t)e * HEADS) = al;

    float* sp = seg_sum + (size_t)d * HEADS;
    unsafeAtomicAdd(sp + 0, al.x);
    unsafeAtomicAdd(sp + 1, al.y);
    unsafeAtomicAdd(sp + 2, al.z);
    unsafeAtomicAdd(sp + 3, al.w);
}

// ---------------------------------------------------------------------------
// 5) Weighted scatter: one wave per edge; lane l owns channels 4l..4l+3
//    (head = l>>3). 512B float4 gather of xt[src], 128 hw f32 atomics to out[dst].
//    Both xt and out fit in the 192MB L2 -> on-chip random access.
// ---------------------------------------------------------------------------
__global__ __launch_bounds__(256) void gat_scatter(const int* __restrict__ ei,
                                                   int E, int N,
                                                   const float* __restrict__ xt,
                                                   const float* __restrict__ alpha,
                                                   const float* __restrict__ seg_sum,
                                                   float* __restrict__ out) {
    int gw   = (blockIdx.x * 256 + threadIdx.x) >> 5;  // edge id
    int lane = threadIdx.x & 31;
    int T = E + N;
    if (gw >= T) return;
    int s, d;
    if (gw < E) { s = ei[gw]; d = ei[E + gw]; } else { s = d = gw - E; }

    int   h     = lane >> 3;
    float w     = alpha[(size_t)gw * HEADS + h];
    float sum   = seg_sum[(size_t)d * HEADS + h];
    float coeff = w / (sum + 1e-16f);

    float4 xv = *(const float4*)(xt + (size_t)s * OUT_CH + lane * 4);
    float* op = out + (size_t)d * OUT_CH + lane * 4;
    unsafeAtomicAdd(op + 0, coeff * xv.x);
    unsafeAtomicAdd(op + 1, coeff * xv.y);
    unsafeAtomicAdd(op + 2, coeff * xv.z);
    unsafeAtomicAdd(op + 3, coeff * xv.w);
}

// ---------------------------------------------------------------------------
// 6) Epilogue: out = relu(out + bias)
// ---------------------------------------------------------------------------
__global__ __launch_bounds__(256) void gat_finalize(float* __restrict__ out,
                                                    const float* __restrict__ bias,
                                                    int N) {
    int i = blockIdx.x * 256 + threadIdx.x;
    if (i >= N * OUT_CH) return;
    float v = out[i] + bias[i & (OUT_CH - 1)];
    out[i] = (v > 0.0f) ? v : 0.0f;
}

// ---------------------------------------------------------------------------
extern "C" void kernel_launch(void* const* d_in, const int* in_sizes, int n_in,
                              void* d_out, int out_size, void* d_ws, size_t ws_size,
                              hipStream_t stream) {
    const float* x       = (const float*)d_in[0];
    const int*   ei      = (const int*)d_in[1];
    // d_in[2] = batch (unused by reference output path)
    const float* W       = (const float*)d_in[3];
    const float* att_src = (const float*)d_in[4];
    const float* att_dst = (const float*)d_in[5];
    const float* bias    = (const float*)d_in[6];
    float*       out     = (float*)d_out;

    const int N = in_sizes[0] / IN_CH;
    const int E = in_sizes[1] / 2;
    const int T = E + N;

    // Workspace carve-up (floats): xt | a_src | a_dst | seg_max | seg_sum | alpha
    float* ws     = (float*)d_ws;
    size_t off    = 0;
    float* xt     = ws + off; off += (size_t)N * OUT_CH;
    float* a_src  = ws + off; off += (size_t)N * HEADS;
    float* a_dst  = ws + off; off += (size_t)N * HEADS;
    float* smax   = ws + off; off += (size_t)N * HEADS;
    float* ssum   = ws + off; off += (size_t)N * HEADS;
    float* alpha  = ws + off; off += (size_t)T * HEADS;
    (void)ws_size; (void)n_in; (void)out_size;

    // 0) init accumulators (fresh every call -> graph-replay safe)
    {
        int total = N * OUT_CH;
        gat_init<<<(total + 255) / 256, 256, 0, stream>>>(out, smax, ssum, N);
    }
    // 1) xt = x @ W (WMMA f32)
    {
        gat_gemm<<<(N + 15) / 16, 256, 0, stream>>>(x, W, xt, N);
    }
    // 2) attention dots (wave per node)
    {
        gat_att_dots<<<(N + 7) / 8, 256, 0, stream>>>(xt, att_src, att_dst,
                                                      a_src, a_dst, N);
    }
    // 3) logits + segment max
    {
        gat_edge_logits<<<(T + 255) / 256, 256, 0, stream>>>(ei, E, N, a_src, a_dst,
                                                             alpha, smax);
    }
    // 4) exp + segment sum
    {
        gat_edge_exp<<<(T + 255) / 256, 256, 0, stream>>>(ei, E, N, smax, alpha, ssum);
    }
    // 5) weighted scatter (wave per edge)
    {
        gat_scatter<<<(T + 7) / 8, 256, 0, stream>>>(ei, E, N, xt, alpha, ssum, out);
    }
    // 6) bias + relu
    {
        int total = N * OUT_CH;
        gat_finalize<<<(total + 255) / 256, 256, 0, stream>>>(out, bias, N);
    }
}